// LinkPredictor_72112500900313
// MI455X (gfx1250) — compile-verified
//
#include <hip/hip_runtime.h>

typedef __attribute__((ext_vector_type(2))) float        v2f;
typedef __attribute__((ext_vector_type(8))) float        v8f;
typedef __attribute__((ext_vector_type(4))) unsigned int v4u;
typedef __attribute__((ext_vector_type(8))) int          v8i;
typedef __attribute__((ext_vector_type(4))) int          v4i;

#define N_NODES 100000
#define D_FEAT  256
#define E_EDGES 320000
#define Q_EDGES 500000
#define M_TILES 6250          // N_NODES/16 exactly
#define LDS_ROW 72            // 64 data floats + 8 pad floats (bank-conflict-free)

// ---------------------------------------------------------------------------
// Phase 0: zero the aggregation buffer (N*D floats = 6.4M float4)
// ---------------------------------------------------------------------------
__global__ __launch_bounds__(256) void gnn_zero_kernel(float4* __restrict__ p) {
    long i = (long)blockIdx.x * blockDim.x + threadIdx.x;   // exactly N*D/4 threads
    p[i] = make_float4(0.f, 0.f, 0.f, 0.f);
}

// ---------------------------------------------------------------------------
// Phase 1: weighted scatter-add.  One wave32 per edge; each lane covers
// elements [4L,4L+4) and [128+4L,128+4L+4) of the 256-wide row, so each
// float4 load is a fully-coalesced 512B burst across the wave.  agg (102MB)
// is L2-resident, so the 8 f32 atomics/lane land in the L2 atomic units.
// ---------------------------------------------------------------------------
__global__ __launch_bounds__(256) void gnn_scatter_kernel(
        const float* __restrict__ x,
        const int*   __restrict__ edge_index,   // [2,E] flat: src row then dst row
        const float* __restrict__ edge_weight,
        float*       __restrict__ agg) {
    int e    = (int)((blockIdx.x * blockDim.x + threadIdx.x) >> 5);   // wave id == edge id
    int lane = threadIdx.x & 31;

    int   src = edge_index[e];
    int   dst = edge_index[E_EDGES + e];
    float w   = edge_weight[e];

    const float4* xs = (const float4*)(x + (long)src * D_FEAT);
    float*        ad = agg + (long)dst * D_FEAT;

    float4 v0 = xs[lane];          // elems 4L..4L+3
    float4 v1 = xs[lane + 32];     // elems 128+4L..

    int o0 = 4 * lane;
    unsafeAtomicAdd(ad + o0 + 0,   w * v0.x);
    unsafeAtomicAdd(ad + o0 + 1,   w * v0.y);
    unsafeAtomicAdd(ad + o0 + 2,   w * v0.z);
    unsafeAtomicAdd(ad + o0 + 3,   w * v0.w);
    unsafeAtomicAdd(ad + o0 + 128, w * v1.x);
    unsafeAtomicAdd(ad + o0 + 129, w * v1.y);
    unsafeAtomicAdd(ad + o0 + 130, w * v1.z);
    unsafeAtomicAdd(ad + o0 + 131, w * v1.w);
}

// ---------------------------------------------------------------------------
// Phase 2: h = agg @ W via V_WMMA_F32_16X16X4_F32.
//
// Block = 8 waves = 8 consecutive 16-row m-tiles, all sharing one 64-column
// strip of W.  The strip (64x256 f32) is staged into LDS once per block by
// the Tensor Data Mover (tensor_load_to_lds, D# per ISA ch.8):
//   tensor: 256x256 f32, stride 256;  tile: dim0=64 (cols), dim1=256 (rows)
//   pad_enable: +8 DWORDs after every 64 DWORDs -> LDS row stride 72 floats,
//   which puts lanes 16-31 at +144 DW = +16 banks from lanes 0-15, so the
//   half-wave B-fragment reads hit disjoint LDS banks.
//
// WMMA fragment layouts per ISA 7.12.2:
//   A 16x4:  lanes 0-15 -> {K=k,k+1}, lanes 16-31 -> {K=k+2,k+3}, M = lane%16
//   B 4x16:  transpose of A layout, N = n0 + lane%16
//   C/D:     VGPR r -> M = r + (lane>=16 ? 8 : 0), N = lane%16
// ---------------------------------------------------------------------------
__global__ __launch_bounds__(256) void gnn_gemm_kernel(
        const float* __restrict__ agg,
        const float* __restrict__ W,
        float*       __restrict__ h) {
    extern __shared__ float Wlds[];                 // 256 * LDS_ROW floats (72KB)

    int quad = blockIdx.x & 3;                      // 64-col strip id
    int mg   = blockIdx.x >> 2;                     // m-tile group (8 tiles)
    int n0   = quad * 64;

    // --- TDM: stage W[:, n0:n0+64] into LDS (issued once, by wave 0) ------
    if (threadIdx.x < 32) {
        unsigned long long ga = (unsigned long long)(const void*)(W + n0);
        unsigned lds_off = (unsigned)(unsigned long long)Wlds;

        v4u g0;
        g0.x = 1u;                                          // count=1, user desc
        g0.y = lds_off;                                     // lds_addr (bytes)
        g0.z = (unsigned)(ga & 0xffffffffu);                // global_addr[31:0]
        g0.w = (unsigned)((ga >> 32) & 0x01ffffffu)         // global_addr[56:32]
             | (2u << 30);                                  // type = 2 ("image")

        v8i g1;
        g1[0] = (int)((2u << 16)                            // data_size = 4B
                    | (1u << 20)                            // pad_enable
                    | (5u << 22)                            // pad_interval = 64 DW
                    | (7u << 25));                          // pad_amount   = 8 DW
        g1[1] = (int)(256u << 16);                          // tensor_dim0 = 256
        g1[2] = (int)(256u << 16);                          // tensor_dim1 = 256
        g1[3] = (int)(64u  << 16);                          // tile_dim0   = 64
        g1[4] = 256;                                        // tile_dim1   = 256
        g1[5] = 256;                                        // tensor_dim0_stride
        g1[6] = 0;
        g1[7] = 0;

        v4i gz = {0, 0, 0, 0};                              // groups 2/3 unused (2D)
#if defined(__clang_major__) && (__clang_major__ >= 23)
        v8i gz8 = {0, 0, 0, 0, 0, 0, 0, 0};
        __builtin_amdgcn_tensor_load_to_lds(g0, g1, gz, gz, gz8, 0);
#else
        __builtin_amdgcn_tensor_load_to_lds(g0, g1, gz, gz, 0);
#endif
        __builtin_amdgcn_s_wait_tensorcnt(0);
    }
    __syncthreads();

    // --- per-wave 16x64 strip of the output -------------------------------
    int wv = threadIdx.x >> 5;
    int mt = mg * 8 + wv;
    if (mt >= M_TILES) return;                      // wave-uniform: EXEC stays full

    int  lane = threadIdx.x & 31;
    int  l16  = lane & 15;
    bool hi   = lane >= 16;
    int  koff = hi ? 2 : 0;

    int m0 = mt * 16;
    const float* aptr = agg + (long)(m0 + l16) * D_FEAT + koff;

    v8f acc0 = {}, acc1 = {}, acc2 = {}, acc3 = {};

    for (int k = 0; k < D_FEAT; k += 4) {
        v2f a;
        a.x = aptr[k];
        a.y = aptr[k + 1];

        const float* wk = &Wlds[(k + koff) * LDS_ROW + l16];
        v2f b0; b0.x = wk[0];  b0.y = wk[LDS_ROW];
        v2f b1; b1.x = wk[16]; b1.y = wk[LDS_ROW + 16];
        v2f b2; b2.x = wk[32]; b2.y = wk[LDS_ROW + 32];
        v2f b3; b3.x = wk[48]; b3.y = wk[LDS_ROW + 48];

        acc0 = __builtin_amdgcn_wmma_f32_16x16x4_f32(false, a, false, b0, (short)0, acc0, false, false);
        acc1 = __builtin_amdgcn_wmma_f32_16x16x4_f32(false, a, false, b1, (short)0, acc1, false, false);
        acc2 = __builtin_amdgcn_wmma_f32_16x16x4_f32(false, a, false, b2, (short)0, acc2, false, false);
        acc3 = __builtin_amdgcn_wmma_f32_16x16x4_f32(false, a, false, b3, (short)0, acc3, false, false);
    }

    int    srow = m0 + (hi ? 8 : 0);
    float* hp   = h + (long)srow * D_FEAT + n0 + l16;
#pragma unroll
    for (int r = 0; r < 8; ++r) {
        hp[(long)r * D_FEAT + 0]  = acc0[r];
        hp[(long)r * D_FEAT + 16] = acc1[r];
        hp[(long)r * D_FEAT + 32] = acc2[r];
        hp[(long)r * D_FEAT + 48] = acc3[r];
    }
}

// ---------------------------------------------------------------------------
// Phase 3: link-prediction decoder.  One wave32 per query edge; two coalesced
// float4 loads per endpoint row (h is L2-resident), butterfly reduction.
// ---------------------------------------------------------------------------
__global__ __launch_bounds__(256) void gnn_decode_kernel(
        const float* __restrict__ h,
        const int*   __restrict__ edges,     // [2,Q] flat
        float*       __restrict__ out) {
    int q    = (int)((blockIdx.x * blockDim.x + threadIdx.x) >> 5);  // wave id == query id
    int lane = threadIdx.x & 31;

    int i = edges[q];
    int j = edges[Q_EDGES + q];

    const float4* hi4 = (const float4*)(h + (long)i * D_FEAT);
    const float4* hj4 = (const float4*)(h + (long)j * D_FEAT);

    float4 a0 = hi4[lane];
    float4 a1 = hi4[lane + 32];
    float4 b0 = hj4[lane];
    float4 b1 = hj4[lane + 32];

    float s = a0.x * b0.x + a0.y * b0.y + a0.z * b0.z + a0.w * b0.w
            + a1.x * b1.x + a1.y * b1.y + a1.z * b1.z + a1.w * b1.w;

#pragma unroll
    for (int off = 16; off > 0; off >>= 1)
        s += __shfl_xor(s, off, 32);

    if (lane == 0) out[q] = s;
}

// ---------------------------------------------------------------------------
extern "C" void kernel_launch(void* const* d_in, const int* in_sizes, int n_in,
                              void* d_out, int out_size, void* d_ws, size_t ws_size,
                              hipStream_t stream) {
    const float* x           = (const float*)d_in[0];   // [N, D]
    const int*   edge_index  = (const int*)  d_in[1];   // [2, E]
    const float* edge_weight = (const float*)d_in[2];   // [E]
    const int*   edges       = (const int*)  d_in[3];   // [2, Q]
    const float* W           = (const float*)d_in[4];   // [D, D]
    float*       out         = (float*)d_out;           // [Q]

    float* agg = (float*)d_ws;                          // [N, D]  (102.4 MB)
    float* h   = agg + (size_t)N_NODES * D_FEAT;        // [N, D]  (102.4 MB)

    // Phase 0: zero agg (6.4M float4 -> 25000 blocks x 256)
    gnn_zero_kernel<<<(N_NODES * D_FEAT / 4) / 256, 256, 0, stream>>>((float4*)agg);

    // Phase 1: scatter-add (one wave per edge -> E/8 blocks of 8 waves)
    gnn_scatter_kernel<<<E_EDGES / 8, 256, 0, stream>>>(x, edge_index, edge_weight, agg);

    // Phase 2: GEMM.  ceil(6250/8)=782 m-groups x 4 col-strips; 72KB dynamic LDS.
    gnn_gemm_kernel<<<782 * 4, 256, 256 * LDS_ROW * 4, stream>>>(agg, W, h);

    // Phase 3: decoder (one wave per query -> Q/8 blocks)
    gnn_decode_kernel<<<Q_EDGES / 8, 256, 0, stream>>>(h, edges, out);
}